// LowRankLayer_Lstep_dilation_75960791597923
// MI455X (gfx1250) — compile-verified
//
#include <hip/hip_runtime.h>

typedef float v2f __attribute__((ext_vector_type(2)));
typedef float v8f __attribute__((ext_vector_type(8)));

#define HH_ 160
#define WW_ 160
#define HWSZ (HH_*WW_)
#define CCH 32
#define KK 9
#define NSTAGES 9
#define BLK 32          // 1 wave per block
#define XSTRIDE 292     // 9*32 = 288 dwords + pad, keeps 16B alignment

// ---------------- head: xh = relu(head_w @ x), stored NHWC ----------------
__global__ __launch_bounds__(256) void head_kernel(
    const float* __restrict__ x,      // [2][3][160][160]
    const float* __restrict__ hwt,    // [32][3]
    float* __restrict__ xh)           // [2][160][160][32]  (NHWC)
{
  int p = blockIdx.x * blockDim.x + threadIdx.x;
  if (p >= 2 * HWSZ) return;
  int b = p / HWSZ, hw = p % HWSZ;
  const float* xb = x + (size_t)b * 3 * HWSZ + hw;
  float x0 = xb[0], x1 = xb[HWSZ], x2 = xb[2 * HWSZ];
  float* o = xh + (size_t)p * CCH;
#pragma unroll
  for (int oc = 0; oc < CCH; ++oc) {
    float v = hwt[oc * 3 + 0] * x0 + hwt[oc * 3 + 1] * x1 + hwt[oc * 3 + 2] * x2;
    o[oc] = fmaxf(v, 0.0f);
  }
}

// ---------------- fused stage loop + WMMA tail ----------------
__global__ __launch_bounds__(BLK) void lowrank_kernel(
    const float* __restrict__ xh,      // [2][160][160][32] NHWC
    const float* __restrict__ tailw,   // [3][32]
    const float* __restrict__ alphas,  // [9]
    const float* __restrict__ betas,   // [9]
    float* __restrict__ out)           // [2][3][160][160]
{
  extern __shared__ float smem[];
  float* Xl  = smem;                    // BLK * XSTRIDE  (per-thread X tile, [k][c])
  float* UVl = smem + BLK * XSTRIDE;    // CCH * BLK      (UVc staged for WMMA B)

  const int t  = threadIdx.x;           // lane (one wave per block)
  const int p  = blockIdx.x * BLK + t;  // global pixel
  const int b  = p / HWSZ;
  const int hw = p % HWSZ;
  const int h  = hw / WW_, w = hw % WW_;

  float* Xt = Xl + t * XSTRIDE;

  float u[CCH];
  float v[KK];
#pragma unroll
  for (int c = 0; c < CCH; ++c) u[c] = 0.0f;

  // ---- gather scrambled-unfold X into LDS; fuse U/V init accumulation ----
#pragma unroll
  for (int k = 0; k < KK; ++k) {
    int m   = h * (WW_ * KK) + w * KK + k;     // flat index of torch view
    int q   = m / HWSZ;                        // kernel-tap index 0..8
    int rem = m - q * HWSZ;
    int hp  = rem / WW_;
    int wp  = rem - hp * WW_;
    int ki  = q / 3;
    int kj  = q - 3 * ki;
    int hs  = hp + ki - 1; hs = hs < 0 ? 0 : (hs > HH_ - 1 ? HH_ - 1 : hs);
    int ws  = wp + kj - 1; ws = ws < 0 ? 0 : (ws > WW_ - 1 ? WW_ - 1 : ws);
    const float4* src = (const float4*)(xh + ((size_t)b * HWSZ + (size_t)hs * WW_ + ws) * CCH);
    float4* dst = (float4*)(Xt + k * CCH);
    float vk = 0.0f;
#pragma unroll
    for (int cc = 0; cc < 8; ++cc) {
      float4 d = src[cc];
      dst[cc] = d;
      u[4 * cc + 0] += d.x; u[4 * cc + 1] += d.y;
      u[4 * cc + 2] += d.z; u[4 * cc + 3] += d.w;
      vk += d.x + d.y + d.z + d.w;
    }
    v[k] = vk * (1.0f / 32.0f);   // V = column mean over C
  }
#pragma unroll
  for (int c = 0; c < CCH; ++c) u[c] *= (1.0f / 9.0f);  // U = row mean over kk

  // ---- 9 stages (rank-collapsed: all 3 rank columns/rows stay identical) ----
  for (int s = 0; s < NSTAGES; ++s) {
    float alpha = alphas[s];
    float beta  = betas[s];

    // pass 1: s2u, numV[k] = u^T X ; V update
    float s2u = 0.0f;
#pragma unroll
    for (int c = 0; c < CCH; ++c) s2u += u[c] * u[c];
    float numV[KK];
#pragma unroll
    for (int k = 0; k < KK; ++k) {
      const float4* row = (const float4*)(Xt + k * CCH);
      float acc = 0.0f;
#pragma unroll
      for (int cc = 0; cc < 8; ++cc) {
        float4 d = row[cc];
        acc += u[4 * cc + 0] * d.x + u[4 * cc + 1] * d.y
             + u[4 * cc + 2] * d.z + u[4 * cc + 3] * d.w;
      }
      numV[k] = acc;
    }
    float c3s2u = 3.0f * s2u;
#pragma unroll
    for (int k = 0; k < KK; ++k)
      v[k] = v[k] - beta * (c3s2u * v[k] - numV[k]);

    // pass 2: s2v, Xv[c] = X v ; U update
    float s2v = 0.0f;
#pragma unroll
    for (int k = 0; k < KK; ++k) s2v += v[k] * v[k];
    float4 Xv[8];
#pragma unroll
    for (int cc = 0; cc < 8; ++cc) Xv[cc] = make_float4(0.f, 0.f, 0.f, 0.f);
#pragma unroll
    for (int k = 0; k < KK; ++k) {
      const float4* row = (const float4*)(Xt + k * CCH);
      float vk = v[k];
#pragma unroll
      for (int cc = 0; cc < 8; ++cc) {
        float4 d = row[cc];
        Xv[cc].x += vk * d.x; Xv[cc].y += vk * d.y;
        Xv[cc].z += vk * d.z; Xv[cc].w += vk * d.w;
      }
    }
    float c3s2v = 3.0f * s2v;
#pragma unroll
    for (int cc = 0; cc < 8; ++cc) {
      u[4 * cc + 0] -= alpha * (c3s2v * u[4 * cc + 0] - Xv[cc].x);
      u[4 * cc + 1] -= alpha * (c3s2v * u[4 * cc + 1] - Xv[cc].y);
      u[4 * cc + 2] -= alpha * (c3s2v * u[4 * cc + 2] - Xv[cc].z);
      u[4 * cc + 3] -= alpha * (c3s2v * u[4 * cc + 3] - Xv[cc].w);
    }
  }

  // ---- UVc[c] = 3 * u[c] * v[center=4]; stage to LDS in WMMA-B layout ----
  float v4c = 3.0f * v[4];
#pragma unroll
  for (int c = 0; c < CCH; ++c) UVl[c * BLK + t] = u[c] * v4c;
  __syncthreads();

  // ---- tail GEMM via fp32 WMMA: D(16x16) = tailw(3x32 pad 16x32) * UVc(32x16) ----
  const int lane = t;
  const int mrow = lane & 15;
  const bool lo  = (lane < 16);       // lanes 0-15: K pair {0,1}; 16-31: {2,3}
  float4 twv[8];
#pragma unroll
  for (int cc = 0; cc < 8; ++cc)
    twv[cc] = (mrow < 3) ? ((const float4*)(tailw + mrow * CCH))[cc]
                         : make_float4(0.f, 0.f, 0.f, 0.f);

  const int pix0   = blockIdx.x * BLK;     // block never crosses batch (25600 % 32 == 0)
  const int bb     = pix0 / HWSZ;
  const int hwbase = pix0 % HWSZ;

#pragma unroll
  for (int half = 0; half < 2; ++half) {
    v8f acc = {0.f, 0.f, 0.f, 0.f, 0.f, 0.f, 0.f, 0.f};
#pragma unroll
    for (int k4 = 0; k4 < 8; ++k4) {         // K = 32 in steps of 4
      float4 q = twv[k4];
      v2f a, bm;
      a.x = lo ? q.x : q.z;                  // A: lane-half selects K pair
      a.y = lo ? q.y : q.w;
      int kk = 4 * k4 + (lo ? 0 : 2);
      bm.x = UVl[(kk + 0) * BLK + half * 16 + mrow];
      bm.y = UVl[(kk + 1) * BLK + half * 16 + mrow];
      acc = __builtin_amdgcn_wmma_f32_16x16x4_f32(
          /*neg_a=*/false, a, /*neg_b=*/false, bm,
          /*c_mod=*/(short)0, acc, /*reuse_a=*/false, /*reuse_b=*/false);
    }
    if (lane < 16) {                          // rows M=0..2 live in lanes 0-15, acc[0..2]
      int hwp = hwbase + half * 16 + lane;
      float* ob = out + (size_t)bb * 3 * HWSZ + hwp;
      ob[0]        = fmaxf(acc[0], 0.0f);
      ob[HWSZ]     = fmaxf(acc[1], 0.0f);
      ob[2 * HWSZ] = fmaxf(acc[2], 0.0f);
    }
  }
}

extern "C" void kernel_launch(void* const* d_in, const int* in_sizes, int n_in,
                              void* d_out, int out_size, void* d_ws, size_t ws_size,
                              hipStream_t stream) {
  const float* x      = (const float*)d_in[0];
  const float* head_w = (const float*)d_in[1];
  const float* tail_w = (const float*)d_in[2];
  const float* alphas = (const float*)d_in[3];
  const float* betas  = (const float*)d_in[4];
  float* out = (float*)d_out;
  float* xh  = (float*)d_ws;   // 2*160*160*32 floats = 6.25 MB (NHWC)

  head_kernel<<<(2 * HWSZ + 255) / 256, 256, 0, stream>>>(x, head_w, xh);

  size_t shmem = (size_t)(BLK * XSTRIDE + CCH * BLK) * sizeof(float);  // 41,472 B
  lowrank_kernel<<<(2 * HWSZ) / BLK, BLK, shmem, stream>>>(xh, tail_w, alphas, betas, out);
}